// GTConv_5111011083066
// MI455X (gfx1250) — compile-verified
//
#include <hip/hip_runtime.h>
#include <hip/hip_bf16.h>
#include <math.h>

// Problem constants (match reference)
constexpr int RR = 8;         // relations (in_channels)
constexpr int CC = 4;         // out_channels
constexpr int EE = 2000000;   // edges per relation

typedef __attribute__((ext_vector_type(2))) float v2f;
typedef __attribute__((ext_vector_type(8))) float v8f;

// ---------------------------------------------------------------------------
// Kernel 1: Filter = softmax(weight, axis=1).  One wave; lanes map to (c, r).
// Writes Filter to d_ws (consumed by scale_kernel) and to the tail of d_out
// (second tuple output of the reference).
// ---------------------------------------------------------------------------
__global__ void GTConv_filter_kernel(const float* __restrict__ weight,
                                     float* __restrict__ filt_ws,
                                     float* __restrict__ out_tail) {
    const int tid = threadIdx.x;          // 0..31
    const int c = tid >> 3;
    const int r = tid & 7;
    float w = weight[c * RR + r];

    // max over the 8 lanes of this c-row (groups of 8 within the wave)
    float m = w;
    #pragma unroll
    for (int off = 4; off > 0; off >>= 1)
        m = fmaxf(m, __shfl_xor(m, off, 8));
    float ex = expf(w - m);
    float s = ex;
    #pragma unroll
    for (int off = 4; off > 0; off >>= 1)
        s += __shfl_xor(s, off, 8);
    float f = ex / s;

    filt_ws[tid]  = f;   // [C,R] row-major: tid == c*R + r
    out_tail[tid] = f;
}

// ---------------------------------------------------------------------------
// Kernel 2: out[c,r,e] = Filter[c,r] * edge_w[r,e] via V_WMMA_F32_16X16X4_F32.
//   M = 16 (c,rs) pairs, K = 4 relations of a half, N = 16 edges.
//   A[(c,rs),k] = (k==rs) ? Filter[c, rbase+rs] : 0   (loop-invariant)
//   B[k,n]      = edge_w[rbase+k, e0+n]
//   D[(c,rs),n] = Filter[c,rbase+rs] * edge_w[rbase+rs, e0+n]  (fully dense)
// Memory-bound: ~320 MB @ 23.3 TB/s => ~14 us roofline. NT hints on both
// streams (no reuse; output 256 MB > 192 MB L2).
// ---------------------------------------------------------------------------
__global__ void GTConv_scale_kernel(const float* __restrict__ edge_w,
                                    const float* __restrict__ filt,
                                    float* __restrict__ out) {
    const int lane = threadIdx.x & 31;
    const int half = lane >> 4;      // 0: lanes 0-15, 1: lanes 16-31
    const int l16  = lane & 15;

    const int waves_per_block = blockDim.x >> 5;
    const int wave   = blockIdx.x * waves_per_block + (threadIdx.x >> 5);
    const int nwaves = gridDim.x * waves_per_block;

    // Build the two diagonal-loaded A operands (relation halves rbase=0,4).
    // A layout (16x4 f32): lanes 0-15 -> M=lane, V0=K0, V1=K1;
    //                      lanes 16-31 -> M=lane-16, V0=K2, V1=K3.
    v2f aH[2];
    {
        const int m    = l16;
        const int cidx = m >> 2;
        const int rs   = m & 3;
        const int k0   = half ? 2 : 0;
        #pragma unroll
        for (int h = 0; h < 2; ++h) {
            const float f = filt[cidx * RR + h * 4 + rs];
            v2f a;
            a.x = (rs == k0    ) ? f : 0.0f;
            a.y = (rs == k0 + 1) ? f : 0.0f;
            aH[h] = a;
        }
    }

    const int ntiles = EE / 64;   // 31250, exact
    for (int t = wave; t < ntiles; t += nwaves) {
        const int e0 = t * 64;
        #pragma unroll
        for (int h = 0; h < 2; ++h) {
            const int rbase = h * 4;
            // B layout mirrors A: V0 holds K rows {0,2}, V1 holds {1,3}.
            const size_t row0 = (size_t)(rbase + (half ? 2 : 0)) * EE;
            const size_t row1 = row0 + EE;
            #pragma unroll
            for (int s = 0; s < 4; ++s) {
                const int e = e0 + s * 16 + l16;
                v2f b;
                b.x = __builtin_nontemporal_load(&edge_w[row0 + e]);
                b.y = __builtin_nontemporal_load(&edge_w[row1 + e]);
                v8f cz = {};
                v8f d = __builtin_amdgcn_wmma_f32_16x16x4_f32(
                    /*neg_a=*/false, aH[h], /*neg_b=*/false, b,
                    /*c_mod=*/(short)0, cz, /*reuse_a=*/false, /*reuse_b=*/false);
                // D layout: lane n<16 holds N=n rows M=v; lanes 16-31 rows M=v+8.
                #pragma unroll
                for (int v = 0; v < 8; ++v) {
                    const int m    = v + (half ? 8 : 0);
                    const int cidx = m >> 2;
                    const int rs   = m & 3;
                    const size_t idx =
                        ((size_t)cidx * RR + (rbase + rs)) * (size_t)EE + (size_t)e;
                    __builtin_nontemporal_store(d[v], &out[idx]);
                }
            }
        }
    }
}

// ---------------------------------------------------------------------------
extern "C" void kernel_launch(void* const* d_in, const int* in_sizes, int n_in,
                              void* d_out, int out_size, void* d_ws, size_t ws_size,
                              hipStream_t stream) {
    const float* edge_w = (const float*)d_in[0];   // [R, E] f32
    const float* weight = (const float*)d_in[1];   // [C, R] f32
    float* out  = (float*)d_out;                   // [C,R,E] then [C,R]
    float* filt = (float*)d_ws;                    // 32 floats scratch

    GTConv_filter_kernel<<<1, 32, 0, stream>>>(
        weight, filt, out + (size_t)CC * RR * EE);

    // 2048 blocks * 8 waves = 16384 waves over 31250 tiles (~2 each).
    GTConv_scale_kernel<<<2048, 256, 0, stream>>>(edge_w, filt, out);
}